// SpatialTransformer2dAffineLayer_36352603193671
// MI455X (gfx1250) — compile-verified
//
#include <hip/hip_runtime.h>
#include <hip/hip_bf16.h>

// ---------------------------------------------------------------------------
// Spatial transformer (affine) for MI455X / gfx1250.
//   Kernel 1: localization head  theta = tanh(theta_feat @ W + b)
//             -> V_WMMA_F32_16X16X4_F32 (one wave, K=128 as 32 k-steps),
//                branch-free B-operand masking (keeps EXEC all-1s region clean)
//   Kernel 2: bilinear resampler (bandwidth-bound; 128-bit gathers,
//             non-temporal 128-bit stores so output doesn't evict input in L2)
// ---------------------------------------------------------------------------

typedef float v2f __attribute__((ext_vector_type(2)));
typedef float v4f __attribute__((ext_vector_type(4)));
typedef float v8f __attribute__((ext_vector_type(8)));

#define IMG_H 256
#define IMG_W 256
#define CHANS 32
#define BATCH 32
#define NFEAT 128

// -------------------------------------------------------------------
// Kernel 1: theta[32,6] = tanh(theta_feat[32,128] @ W[128,6] + b[6])
// One wave (32 lanes). Two 16x16 accumulator tiles cover M=0..31.
// A layout (16x4 f32, ISA 7.12.2): lanes 0-15 -> K=0,1 ; lanes 16-31 -> K=2,3
// B layout (4x16): N striped across lanes, same K split.
// C/D layout: VGPR j, lanes 0-15 -> M=j, N=lane%16 ; lanes 16-31 -> M=j+8.
// -------------------------------------------------------------------
__global__ void __launch_bounds__(32)
stn_theta_wmma_kernel(const float* __restrict__ tf,   // [32,128]
                      const float* __restrict__ W,    // [128,6]
                      const float* __restrict__ bias, // [6]
                      float* __restrict__ theta)      // [32,6] (d_ws)
{
    const int lane = threadIdx.x;        // 0..31
    const int half = lane >> 4;          // 0 or 1  -> K offset 2*half
    const int l16  = lane & 15;

    // Branch-free B-operand addressing: clamp column, zero via mask.
    const int   wcol  = (l16 < 6) ? l16 : 5;     // safe index
    const float wmask = (l16 < 6) ? 1.0f : 0.0f;

    v8f acc0 = {};   // rows 0..15
    v8f acc1 = {};   // rows 16..31

    for (int k0 = 0; k0 < NFEAT; k0 += 4) {
        const int ka = k0 + 2 * half;

        v2f a0, a1, bb;
        a0.x = tf[l16 * NFEAT + ka];
        a0.y = tf[l16 * NFEAT + ka + 1];
        a1.x = tf[(16 + l16) * NFEAT + ka];
        a1.y = tf[(16 + l16) * NFEAT + ka + 1];

        bb.x = W[ka * 6 + wcol]       * wmask;
        bb.y = W[(ka + 1) * 6 + wcol] * wmask;

        acc0 = __builtin_amdgcn_wmma_f32_16x16x4_f32(
                   false, a0, false, bb, (short)0, acc0, false, false);
        acc1 = __builtin_amdgcn_wmma_f32_16x16x4_f32(
                   false, a1, false, bb, (short)0, acc1, false, false);
    }

    if (l16 < 6) {
        const float bv = bias[l16];
        #pragma unroll
        for (int j = 0; j < 8; ++j) {
            const int m0 = j + 8 * half;         // tile 0 row
            theta[m0 * 6 + l16] = tanhf(acc0[j] + bv);
            const int m1 = 16 + j + 8 * half;    // tile 1 row
            theta[m1 * 6 + l16] = tanhf(acc1[j] + bv);
        }
    }
}

// -------------------------------------------------------------------
// Kernel 2: bilinear resample. One thread per (pixel, 4-channel chunk).
// tid -> (b, oy, ox, chunk); stores are fully coalesced non-temporal
// 128-bit vectors; the 4 gather taps per wave are contiguous 128B cache
// lines with heavy L2 reuse between neighboring output pixels.
// -------------------------------------------------------------------
__global__ void __launch_bounds__(256)
stn_sample_kernel(const float* __restrict__ U,      // [32,256,256,32]
                  const float* __restrict__ theta,  // [32,6]
                  float* __restrict__ out)          // [32,256,256,32]
{
    const int tid   = blockIdx.x * 256 + threadIdx.x;
    const int chunk = tid & 7;           // which 4-channel group of 32 channels
    const int pix   = tid >> 3;
    const int ox    = pix & (IMG_W - 1);
    const int oy    = (pix >> 8) & (IMG_H - 1);
    const int b     = pix >> 16;

    const float* th = theta + b * 6;
    const float t0 = th[0], t1 = th[1], t2 = th[2];
    const float t3 = th[3], t4 = th[4], t5 = th[5];

    // jnp.linspace(-1, 1, 256): step = 2/255
    const float xt = -1.0f + (2.0f / 255.0f) * (float)ox;
    const float yt = -1.0f + (2.0f / 255.0f) * (float)oy;

    const float xs = t0 * xt + t1 * yt + t2;
    const float ys = t3 * xt + t4 * yt + t5;

    // NOTE: reference scales by w (=256), not w-1
    const float x = (xs + 1.0f) * (0.5f * (float)IMG_W);
    const float y = (ys + 1.0f) * (0.5f * (float)IMG_H);

    int x0 = (int)floorf(x);
    int y0 = (int)floorf(y);
    int x1 = x0 + 1;
    int y1 = y0 + 1;
    x0 = min(max(x0, 0), IMG_W - 1);
    x1 = min(max(x1, 0), IMG_W - 1);
    y0 = min(max(y0, 0), IMG_H - 1);
    y1 = min(max(y1, 0), IMG_H - 1);

    const float x0f = (float)x0, x1f = (float)x1;
    const float y0f = (float)y0, y1f = (float)y1;

    const float wa = (x1f - x) * (y1f - y);
    const float wb = (x1f - x) * (y - y0f);
    const float wc = (x - x0f) * (y1f - y);
    const float wd = (x - x0f) * (y - y0f);

    const v4f* base =
        (const v4f*)U + (size_t)b * (IMG_H * IMG_W * (CHANS / 4));

    const v4f Ia = base[(y0 * IMG_W + x0) * (CHANS / 4) + chunk];
    const v4f Ib = base[(y1 * IMG_W + x0) * (CHANS / 4) + chunk];
    const v4f Ic = base[(y0 * IMG_W + x1) * (CHANS / 4) + chunk];
    const v4f Id = base[(y1 * IMG_W + x1) * (CHANS / 4) + chunk];

    const v4f r = wa * Ia + wb * Ib + wc * Ic + wd * Id;

    // Output is write-once/stream: non-temporal store keeps L2 for the input.
    __builtin_nontemporal_store(r, &((v4f*)out)[tid]);
}

extern "C" void kernel_launch(void* const* d_in, const int* in_sizes, int n_in,
                              void* d_out, int out_size, void* d_ws, size_t ws_size,
                              hipStream_t stream) {
    const float* U    = (const float*)d_in[0];  // [32,256,256,32]
    const float* tf   = (const float*)d_in[1];  // [32,128]
    const float* W    = (const float*)d_in[2];  // [128,6]
    const float* bias = (const float*)d_in[3];  // [6]
    float* out   = (float*)d_out;
    float* theta = (float*)d_ws;                // [32,6] scratch

    // 1) localization head via WMMA (one wave)
    stn_theta_wmma_kernel<<<1, 32, 0, stream>>>(tf, W, bias, theta);

    // 2) bilinear resampler: 32*256*256 pixels * 8 four-channel chunks
    const int total_threads = BATCH * IMG_H * IMG_W * (CHANS / 4); // 16,777,216
    stn_sample_kernel<<<total_threads / 256, 256, 0, stream>>>(U, theta, out);
}